// Model_52518860096357
// MI455X (gfx1250) — compile-verified
//
#include <hip/hip_runtime.h>
#include <hip/hip_bf16.h>
#include <math.h>

// Problem constants (match reference)
#define Bb   16
#define Ss   2048
#define Hh   15
#define HP   16      // hidden padded to 16 for WMMA
#define OUTD 4
#define SR   36      // LDS tile row stride in floats (144B: 16B aligned, bank-spread)

typedef float v2f __attribute__((ext_vector_type(2)));
typedef float v4f __attribute__((ext_vector_type(4)));
typedef float v8f __attribute__((ext_vector_type(8)));

// ---------------------------------------------------------------------------
// Kernel 1: wq/wk/wv = x @ W.T + b, written zero-padded to HP=16 columns.
// ---------------------------------------------------------------------------
__global__ __launch_bounds__(256) void proj_kernel(
    const float* __restrict__ q, const float* __restrict__ k, const float* __restrict__ v,
    const float* __restrict__ Wq, const float* __restrict__ bq,
    const float* __restrict__ Wk, const float* __restrict__ bk,
    const float* __restrict__ Wv, const float* __restrict__ bv,
    float* __restrict__ wq, float* __restrict__ wk, float* __restrict__ wv)
{
  const int NR = Bb * Ss;
  int idx = blockIdx.x * blockDim.x + threadIdx.x;
  if (idx >= 3 * NR) return;
  int t  = idx / NR;
  int rs = idx - t * NR;
  const float *in, *W, *bias; float* out;
  if (t == 0)      { in = q; W = Wq; bias = bq; out = wq; }
  else if (t == 1) { in = k; W = Wk; bias = bk; out = wk; }
  else             { in = v; W = Wv; bias = bv; out = wv; }
  float x[Hh];
#pragma unroll
  for (int i = 0; i < Hh; ++i) x[i] = in[(size_t)rs * Hh + i];
#pragma unroll
  for (int o = 0; o < HP; ++o) {
    float a = 0.0f;
    if (o < Hh) {
      a = bias[o];
#pragma unroll
      for (int i = 0; i < Hh; ++i) a = fmaf(x[i], W[o * Hh + i], a);
    }
    out[(size_t)rs * HP + o] = a;   // column 15 written as exact 0
  }
}

// ---------------------------------------------------------------------------
// Kernel 2: flash-style attention, 32 keys / iteration, row-owned softmax.
// One wave32 owns one (batch, 16-query tile).  8 waves / 256-thread block.
//   - scores: 2 chains of 4x v_wmma_f32_16x16x4_f32 (fp32-exact, K padded)
//   - softmax: lane L owns query row L&15, columns half*16..+15 (via LDS),
//     only two shfl_xor(...,16) per 32 keys; mask/dropout loads are 64B
//     contiguous per lane (dropout non-temporal: 268MB streamed once)
//   - P -> LDS -> A-layout (ds_load_b64), two P*V WMMA chains
// ---------------------------------------------------------------------------
__global__ __launch_bounds__(256) void attn_kernel(
    const float* __restrict__ wq, const float* __restrict__ wk, const float* __restrict__ wv,
    const float* __restrict__ attn_mask, const float* __restrict__ dropout_mask,
    const float* __restrict__ W1, const float* __restrict__ b1,
    float* __restrict__ out)
{
  // per wave: 16x32 score/P tile (stride SR) + 16-float broadcast buffer
  __shared__ float lds[8][16 * SR + 16];

  const int lane = threadIdx.x & 31;
  const int wid  = threadIdx.x >> 5;
  const int gw   = blockIdx.x * 8 + wid;     // global wave id
  const int b    = gw / (Ss / 16);
  const int qt   = gw - b * (Ss / 16);
  const int qbase = qt * 16;
  const int half = lane >> 4;                // 0 or 1
  const int l16  = lane & 15;

  float* tile   = &lds[wid][0];
  float* bcast  = tile + 16 * SR;            // 16-float row-broadcast buffer

  // ---- Q tile in A layout: lane covers M=l16, K = 4c + 2*half + {0,1} ----
  const float* wqrow = wq + ((size_t)b * Ss + qbase + l16) * HP;
  v2f aq[4];
#pragma unroll
  for (int c = 0; c < 4; ++c)
    aq[c] = *(const v2f*)(wqrow + 4 * c + 2 * half);

  // acc: C layout (vgpr r -> row r + 8*half).  m/Z: per-lane row l16 stats.
  v8f acc = {};
#pragma unroll
  for (int r = 0; r < 8; ++r) acc[r] = 0.0f;
  float m_st = -INFINITY, Z_st = 0.0f;

  const float* mask_row = attn_mask    + (size_t)(qbase + l16) * Ss + 16 * half;
  const float* drop_row = dropout_mask + ((size_t)b * Ss + qbase + l16) * Ss + 16 * half;

  for (int kb = 0; kb < Ss; kb += 32) {
    // ---- K sub-tiles (t=0,1) in B layout, b64 loads ----
    v8f s[2];
#pragma unroll
    for (int t = 0; t < 2; ++t) {
      const float* wkrow = wk + ((size_t)b * Ss + kb + 16 * t + l16) * HP;
      v8f sv = {};
#pragma unroll
      for (int r = 0; r < 8; ++r) sv[r] = 0.0f;
#pragma unroll
      for (int c = 0; c < 4; ++c) {
        v2f bkv = *(const v2f*)(wkrow + 4 * c + 2 * half);
        sv = __builtin_amdgcn_wmma_f32_16x16x4_f32(false, aq[c], false, bkv,
                                                   (short)0, sv, false, false);
      }
      s[t] = sv;
    }

    // ---- scatter C-layout scores into LDS tile [16][32] ----
#pragma unroll
    for (int r = 0; r < 8; ++r) {
      tile[(r + 8 * half) * SR + l16]      = s[0][r];
      tile[(r + 8 * half) * SR + 16 + l16] = s[1][r];
    }

    // ---- row phase: lane owns (query row l16, key cols 16*half..+15) ----
    float x[16];
    {
      const v4f* srow = (const v4f*)(tile + l16 * SR + 16 * half);
      const v4f* mrow = (const v4f*)(mask_row + kb);
#pragma unroll
      for (int i = 0; i < 4; ++i) {
        v4f xv = srow[i];
        v4f mv = mrow[i];
        x[4 * i + 0] = xv.x + mv.x;  x[4 * i + 1] = xv.y + mv.y;
        x[4 * i + 2] = xv.z + mv.z;  x[4 * i + 3] = xv.w + mv.w;
      }
    }
    float rmax = x[0];
#pragma unroll
    for (int i = 1; i < 16; ++i) rmax = fmaxf(rmax, x[i]);
    rmax = fmaxf(rmax, __shfl_xor(rmax, 16, 32));         // combine halves
    const float mnew = fmaxf(m_st, rmax);
    const float scl  = __expf(m_st - mnew);
    m_st = mnew;

    float p[16], psum = 0.0f;
#pragma unroll
    for (int i = 0; i < 16; ++i) { p[i] = __expf(x[i] - mnew); psum += p[i]; }
    psum += __shfl_xor(psum, 16, 32);                     // combine halves
    Z_st = Z_st * scl + psum;                             // Z excludes dropout

    // dropout (streamed once, 64B contiguous per lane, non-temporal)
    {
      const v4f* drow = (const v4f*)(drop_row + kb);
      v4f pdv[4];
#pragma unroll
      for (int i = 0; i < 4; ++i) {
        v4f dv = __builtin_nontemporal_load(drow + i);
        pdv[i].x = p[4 * i + 0] * dv.x;  pdv[i].y = p[4 * i + 1] * dv.y;
        pdv[i].z = p[4 * i + 2] * dv.z;  pdv[i].w = p[4 * i + 3] * dv.w;
      }
      v4f* prow = (v4f*)(tile + l16 * SR + 16 * half);
#pragma unroll
      for (int i = 0; i < 4; ++i) prow[i] = pdv[i];
    }

    // ---- broadcast row rescale factors to C-layout accumulator ----
    bcast[l16] = scl;                 // both halves write identical value
    {
      const float* scrow = bcast + 8 * half;
#pragma unroll
      for (int r = 0; r < 8; ++r) acc[r] *= scrow[r];
    }

    // ---- P (A layout, ds_load_b64) and V (B layout), two WMMA chains ----
#pragma unroll
    for (int t = 0; t < 2; ++t) {
#pragma unroll
      for (int c = 0; c < 4; ++c) {
        v2f pa = *(const v2f*)(tile + l16 * SR + 16 * t + 4 * c + 2 * half);
        const float* vr = wv + ((size_t)b * Ss + kb + 16 * t + 4 * c + 2 * half) * HP;
        v2f bvv;
        bvv.x = vr[l16];
        bvv.y = vr[HP + l16];
        acc = __builtin_amdgcn_wmma_f32_16x16x4_f32(false, pa, false, bvv,
                                                    (short)0, acc, false, false);
      }
    }
  }

  // ---- redistribute Z to C layout, normalize, park out tile in LDS ----
  bcast[l16] = Z_st;
  {
    const float* zrow = bcast + 8 * half;
#pragma unroll
    for (int r = 0; r < 8; ++r)
      tile[(r + 8 * half) * SR + l16] = acc[r] / zrow[r];
  }

  // ---- final projection to OUT=4: lane does row l16, outs {half, half+2} ----
#pragma unroll
  for (int oo = 0; oo < 2; ++oo) {
    const int o = half + 2 * oo;
    float a = b1[o];
#pragma unroll
    for (int h = 0; h < Hh; ++h)
      a = fmaf(tile[l16 * SR + h], W1[o * Hh + h], a);
    out[((size_t)b * Ss + qbase + l16) * OUTD + o] = a;
  }
}

// ---------------------------------------------------------------------------
extern "C" void kernel_launch(void* const* d_in, const int* in_sizes, int n_in,
                              void* d_out, int out_size, void* d_ws, size_t ws_size,
                              hipStream_t stream) {
  // setup_inputs order: q,k,v,attn_mask,dropout_mask,Wq,bq,Wk,bk,Wv,bv,W1,b1
  const float* q            = (const float*)d_in[0];
  const float* k            = (const float*)d_in[1];
  const float* v            = (const float*)d_in[2];
  const float* attn_mask    = (const float*)d_in[3];
  const float* dropout_mask = (const float*)d_in[4];
  const float* Wq = (const float*)d_in[5];
  const float* bq = (const float*)d_in[6];
  const float* Wk = (const float*)d_in[7];
  const float* bk = (const float*)d_in[8];
  const float* Wv = (const float*)d_in[9];
  const float* bv = (const float*)d_in[10];
  const float* W1 = (const float*)d_in[11];
  const float* b1 = (const float*)d_in[12];
  float* out = (float*)d_out;

  const size_t NR = (size_t)Bb * Ss;          // rows per tensor
  float* wsf = (float*)d_ws;                  // 3 * NR * HP floats = 6.3 MB
  float* wq = wsf;
  float* wk = wsf + NR * HP;
  float* wv = wsf + 2 * NR * HP;

  {
    int total = 3 * (int)NR;
    int blocks = (total + 255) / 256;
    proj_kernel<<<blocks, 256, 0, stream>>>(q, k, v, Wq, bq, Wk, bk, Wv, bv,
                                            wq, wk, wv);
  }
  {
    // one wave per (b, 16-query tile): 16 * 128 = 2048 waves, 8 waves/block
    int blocks = (Bb * (Ss / 16)) / 8;        // 256 blocks
    attn_kernel<<<blocks, 256, 0, stream>>>(wq, wk, wv, attn_mask, dropout_mask,
                                            W1, b1, out);
  }
}